// FHNDynamics_21354577396276
// MI455X (gfx1250) — compile-verified
//
#include <hip/hip_runtime.h>

#define ROW_LEN   2048
#define BLOCK     256
#define TAU_F     12.5f
#define THR_F     0.5f

// ---------------------------------------------------------------------------
// gfx1250 async global->LDS copy (ASYNCcnt path, CDNA5 ISA)
// GV mode: vdst = per-lane LDS byte offset, vaddr = per-lane 64-bit address.
// ---------------------------------------------------------------------------
__device__ __forceinline__ void async_load_b128(unsigned lds_off, const void* gaddr) {
    asm volatile("global_load_async_to_lds_b128 %0, %1, off"
                 :
                 : "v"(lds_off), "v"(gaddr)
                 : "memory");
}

__device__ __forceinline__ void wait_asynccnt0() {
    asm volatile("s_wait_asynccnt 0" ::: "memory");
}

__device__ __forceinline__ float clamp3(float x) {
    return fminf(fmaxf(x, -3.0f), 3.0f);   // lowers to v_med3-style min/max pair
}

// Fully-unrolled FHN IMEX update for the common static n_steps.
template <int NS>
__device__ __forceinline__ float fhn_elem_unrolled(float x, float rscale, float a,
                                                   float dt, float alpha, float rden) {
    float s    = x * rscale;
    float z    = (fabsf(x) - THR_F) * 10.0f;
    float gate = 1.0f / (1.0f + __expf(-z));
    float I    = s * (0.1f + 0.9f * gate);
    float v = 0.0f, w = 0.0f;
#pragma unroll
    for (int k = 0; k < NS; ++k) {
        float dv = v - (v * v * v) * (1.0f / 3.0f) - w + I;
        float vn = v + dt * dv;
        float wn = (w + alpha * (vn + a)) * rden;
        v = clamp3(vn);
        w = clamp3(wn);
    }
    return v;
}

__device__ __forceinline__ float fhn_elem_dyn(float x, float rscale, float a,
                                              float dt, float alpha, float rden, int ns) {
    float s    = x * rscale;
    float z    = (fabsf(x) - THR_F) * 10.0f;
    float gate = 1.0f / (1.0f + __expf(-z));
    float I    = s * (0.1f + 0.9f * gate);
    float v = 0.0f, w = 0.0f;
    for (int k = 0; k < ns; ++k) {
        float dv = v - (v * v * v) * (1.0f / 3.0f) - w + I;
        float vn = v + dt * dv;
        float wn = (w + alpha * (vn + a)) * rden;
        v = clamp3(vn);
        w = clamp3(wn);
    }
    return v;
}

__global__ __launch_bounds__(BLOCK) void fhn_kernel(
    const float* __restrict__ stim,
    const float* __restrict__ a_p,
    const float* __restrict__ b_p,
    const float* __restrict__ dt_p,
    const int*   __restrict__ ns_p,
    float* __restrict__ out_resp,
    float* __restrict__ out_v)
{
    __shared__ float srow[ROW_LEN];   // one 8 KB stimulus row
    __shared__ float sred[BLOCK / 32];

    const int      tid = threadIdx.x;
    const size_t   row = (size_t)blockIdx.x;
    const float* __restrict__ rowp = stim + row * ROW_LEN;

    // ---- Stage the row into LDS with async b128 loads (2 x 16B per lane) ----
    // addrspacecast(local->generic) puts the LDS byte offset in the low 32 bits.
    const unsigned lds_base = (unsigned)(size_t)(&srow[0]);
    async_load_b128(lds_base + (unsigned)tid * 16u,           rowp + tid * 4);
    async_load_b128(lds_base + 4096u + (unsigned)tid * 16u,   rowp + 1024 + tid * 4);
    wait_asynccnt0();          // this wave's async loads landed in LDS
    __syncthreads();           // all 8 waves' loads visible workgroup-wide

    // ---- Row max(|x|) reduction: 8 elems/lane -> wave32 shfl_xor -> LDS ----
    float4 x0 = *(const float4*)&srow[tid * 4];
    float4 x1 = *(const float4*)&srow[1024 + tid * 4];

    float m = fmaxf(fmaxf(fmaxf(fabsf(x0.x), fabsf(x0.y)),
                          fmaxf(fabsf(x0.z), fabsf(x0.w))),
                    fmaxf(fmaxf(fabsf(x1.x), fabsf(x1.y)),
                          fmaxf(fabsf(x1.z), fabsf(x1.w))));
#pragma unroll
    for (int off = 16; off > 0; off >>= 1)
        m = fmaxf(m, __shfl_xor(m, off, 32));
    if ((tid & 31) == 0) sred[tid >> 5] = m;
    __syncthreads();

    float mx = fmaxf(fmaxf(fmaxf(sred[0], sred[1]), fmaxf(sred[2], sred[3])),
                     fmaxf(fmaxf(sred[4], sred[5]), fmaxf(sred[6], sred[7])));
    const float scale  = fmaxf(mx, 1e-6f);
    const float rscale = 1.0f / scale;

    // ---- Scalar parameters ----
    const float a     = a_p[0];
    const float b     = b_p[0];
    const float dt    = dt_p[0];
    const int   ns    = ns_p[0];
    const float alpha = dt / TAU_F;
    const float rden  = 1.0f / (1.0f + alpha * b);

    float xs[8] = { x0.x, x0.y, x0.z, x0.w, x1.x, x1.y, x1.z, x1.w };
    float vs[8];

    if (ns == 8) {   // static n_steps in the reference -> fully unrolled fast path
#pragma unroll
        for (int i = 0; i < 8; ++i)
            vs[i] = fhn_elem_unrolled<8>(xs[i], rscale, a, dt, alpha, rden);
    } else {
#pragma unroll
        for (int i = 0; i < 8; ++i)
            vs[i] = fhn_elem_dyn(xs[i], rscale, a, dt, alpha, rden, ns);
    }

    // ---- Coalesced b128 stores for both outputs ----
    const size_t base = row * ROW_LEN;
    float4 r0 = make_float4(vs[0] * scale, vs[1] * scale, vs[2] * scale, vs[3] * scale);
    float4 r1 = make_float4(vs[4] * scale, vs[5] * scale, vs[6] * scale, vs[7] * scale);
    float4 v0 = make_float4(vs[0], vs[1], vs[2], vs[3]);
    float4 v1 = make_float4(vs[4], vs[5], vs[6], vs[7]);

    *(float4*)&out_resp[base + tid * 4]        = r0;
    *(float4*)&out_resp[base + 1024 + tid * 4] = r1;
    *(float4*)&out_v   [base + tid * 4]        = v0;
    *(float4*)&out_v   [base + 1024 + tid * 4] = v1;
}

extern "C" void kernel_launch(void* const* d_in, const int* in_sizes, int n_in,
                              void* d_out, int out_size, void* d_ws, size_t ws_size,
                              hipStream_t stream) {
    const float* stim = (const float*)d_in[0];
    const float* a_p  = (const float*)d_in[1];
    const float* b_p  = (const float*)d_in[2];
    const float* dt_p = (const float*)d_in[3];
    const int*   ns_p = (const int*)d_in[4];

    const size_t total = (size_t)in_sizes[0];        // 4*4096*2048
    const int    rows  = (int)(total / ROW_LEN);     // 16384

    float* out_resp = (float*)d_out;                 // tuple output: (response, v)
    float* out_v    = out_resp + total;

    fhn_kernel<<<dim3(rows), dim3(BLOCK), 0, stream>>>(
        stim, a_p, b_p, dt_p, ns_p, out_resp, out_v);
}